// SGConvolution_6279242186783
// MI455X (gfx1250) — compile-verified
//
#include <hip/hip_runtime.h>
#include <stdint.h>

// ---------------- configuration ----------------
#define TILE            128               // edges per LDS tile (per wave)
#define TILES_PER_CHUNK 2                 // chunk = 256 edges per wave
#define CHUNK           (TILE * TILES_PER_CHUNK)
#define WAVES_PER_BLOCK 8
#define BLOCK_T         (WAVES_PER_BLOCK * 32)
#define DF              64                // feature dim (fixed by reference)

// ---------------- CDNA5 async global->LDS path (probe-guarded) ----------------
#if __has_builtin(__builtin_amdgcn_global_load_async_to_lds_b128)
#define HAVE_ASYNC 1
#else
#define HAVE_ASYNC 0
#endif

// Builtin signature (confirmed by clang diagnostic in an earlier round):
//   void __builtin_amdgcn_global_load_async_to_lds_b128(
//       v4i AS1* gsrc, v4i AS3* lds, imm int offset, imm int cpol)
typedef int v4i __attribute__((vector_size(16)));
typedef __attribute__((address_space(1))) v4i* gas1_v4i_p;
typedef __attribute__((address_space(3))) v4i* las3_v4i_p;

__device__ __forceinline__ void compiler_mem_fence() {
    asm volatile("" ::: "memory");
}

template <int N>
__device__ __forceinline__ void wait_async_le() {
#if HAVE_ASYNC
#if __has_builtin(__builtin_amdgcn_s_wait_asynccnt)
    __builtin_amdgcn_s_wait_asynccnt(N);
#else
    asm volatile("s_wait_asynccnt %0" ::"i"(N) : "memory");
#endif
#endif
}

// ---------------- one SpMM pass: out += A * src (out must be pre-zeroed) --------
// Placed FIRST in the file so the disasm snippet shows this kernel.
__global__ __launch_bounds__(BLOCK_T) void sg_spmm_kernel(
    const float* __restrict__ src,      // [N, 64]
    const int*   __restrict__ erow,     // [E], sorted ascending
    const int*   __restrict__ ecol,     // [E]
    const float* __restrict__ evl,      // [E]
    float*       __restrict__ out,      // [N, 64], zero-initialized
    int n_edges) {
    // double-buffered per-wave edge tiles
    __shared__ int   s_row[2][WAVES_PER_BLOCK][TILE];
    __shared__ int   s_col[2][WAVES_PER_BLOCK][TILE];
    __shared__ float s_val[2][WAVES_PER_BLOCK][TILE];

    const int lane = threadIdx.x & 31;
    const int wv   = threadIdx.x >> 5;

    const int chunk_base = (blockIdx.x * WAVES_PER_BLOCK + wv) * CHUNK;
    if (chunk_base >= n_edges) return;   // whole wave exits; no block barriers used

    const int remaining = n_edges - chunk_base;
    int ntiles = (remaining + TILE - 1) / TILE;
    if (ntiles > TILES_PER_CHUNK) ntiles = TILES_PER_CHUNK;

    // Stream one 128-edge tile (row/col/val) into this wave's LDS slice.
    auto issue_tile = [&](int t, int buf) {
        const int base = chunk_base + t * TILE;
        if (base + TILE <= n_edges) {
            const int off = lane * 4;                       // 4 edges per lane, b128 each
#if HAVE_ASYNC
            __builtin_amdgcn_global_load_async_to_lds_b128(
                (gas1_v4i_p)(erow + base + off),
                (las3_v4i_p)&s_row[buf][wv][off], 0, 0);
            __builtin_amdgcn_global_load_async_to_lds_b128(
                (gas1_v4i_p)(ecol + base + off),
                (las3_v4i_p)&s_col[buf][wv][off], 0, 0);
            __builtin_amdgcn_global_load_async_to_lds_b128(
                (gas1_v4i_p)(evl + base + off),
                (las3_v4i_p)&s_val[buf][wv][off], 0, 0);
#else
            *(int4*)  &s_row[buf][wv][off] = *(const int4*)  (erow + base + off);
            *(int4*)  &s_col[buf][wv][off] = *(const int4*)  (ecol + base + off);
            *(float4*)&s_val[buf][wv][off] = *(const float4*)(evl  + base + off);
#endif
        } else {
            // tail tile: bounded scalar fills with benign sentinels
            for (int j = lane; j < TILE; j += 32) {
                const int e = base + j;
                const bool ok = (e < n_edges);
                s_row[buf][wv][j] = ok ? erow[e] : -1;
                s_col[buf][wv][j] = ok ? ecol[e] : 0;
                s_val[buf][wv][j] = ok ? evl[e]  : 0.0f;
            }
        }
    };

    issue_tile(0, 0);

    float ax = 0.0f, ay = 0.0f;   // per-row accumulator: features [2*lane, 2*lane+1]
    int   cur = -1;

    for (int t = 0; t < ntiles; ++t) {
        const int buf = t & 1;
        // prefetch next tile into the other buffer, then wait for current tile
        const bool have_next  = (t + 1) < ntiles;
        const bool next_async =
            have_next && (chunk_base + (t + 2) * TILE <= n_edges);
        if (have_next) issue_tile(t + 1, buf ^ 1);
        if (next_async) wait_async_le<3>();   // current tile's 3 asyncs retired
        else            wait_async_le<0>();   // drain everything
        __builtin_amdgcn_wave_barrier();
        compiler_mem_fence();

        const int*   rp = s_row[buf][wv];
        const int*   cp = s_col[buf][wv];
        const float* vp = s_val[buf][wv];

        for (int k = 0; k < TILE; ++k) {
            const int   r = rp[k];          // wave-uniform LDS broadcast reads
            const int   c = cp[k];
            const float v = vp[k];
            if (k + 8 < TILE) {             // warm the gather path ahead of time
                __builtin_prefetch(
                    (const char*)(src + (size_t)cp[k + 8] * DF) + lane * 8, 0, 3);
            }
            const float2 g = ((const float2*)(src + (size_t)c * DF))[lane];
            if (r != cur) {                 // rows sorted: flush only on row change
                if (cur >= 0) {
                    float* o = out + (size_t)cur * DF + lane * 2;
                    unsafeAtomicAdd(o + 0, ax);
                    unsafeAtomicAdd(o + 1, ay);
                }
                cur = r; ax = 0.0f; ay = 0.0f;
            }
            ax = fmaf(v, g.x, ax);
            ay = fmaf(v, g.y, ay);
        }
        compiler_mem_fence();
        __builtin_amdgcn_wave_barrier();
    }

    if (cur >= 0) {
        float* o = out + (size_t)cur * DF + lane * 2;
        unsafeAtomicAdd(o + 0, ax);
        unsafeAtomicAdd(o + 1, ay);
    }
}

// ---------------- zero-init for scatter targets ----------------
__global__ void sg_zero2_kernel(float* __restrict__ a, float* __restrict__ b,
                                long long n /* elements, multiple of 4 */) {
    long long i      = (long long)blockIdx.x * blockDim.x + threadIdx.x;
    long long stride = (long long)gridDim.x * blockDim.x;
    long long n4     = n >> 2;
    float4 z; z.x = 0.f; z.y = 0.f; z.z = 0.f; z.w = 0.f;
    for (; i < n4; i += stride) {
        ((float4*)a)[i] = z;
        ((float4*)b)[i] = z;
    }
}

// ---------------- launcher ----------------
extern "C" void kernel_launch(void* const* d_in, const int* in_sizes, int n_in,
                              void* d_out, int out_size, void* d_ws, size_t ws_size,
                              hipStream_t stream) {
    const float* x    = (const float*)d_in[0];   // [N,64] f32
    const int*   erow = (const int*)  d_in[1];   // [E] i32 (sorted)
    const int*   ecol = (const int*)  d_in[2];   // [E] i32
    const float* evl  = (const float*)d_in[3];   // [E] f32

    float* out = (float*)d_out;                  // [N,64] f32
    float* h1  = (float*)d_ws;                   // intermediate [N,64] f32

    const long long n_elems = (long long)out_size;   // N*64
    const int E             = in_sizes[1];

    // zero both scatter targets every call (deterministic across graph replays)
    sg_zero2_kernel<<<1024, 256, 0, stream>>>(h1, out, n_elems);

    const int n_chunks = (E + CHUNK - 1) / CHUNK;
    const int nblocks  = (n_chunks + WAVES_PER_BLOCK - 1) / WAVES_PER_BLOCK;

    // order-2 SGC: h1 = A x ; out = A h1
    sg_spmm_kernel<<<nblocks, BLOCK_T, 0, stream>>>(x,  erow, ecol, evl, h1,  E);
    sg_spmm_kernel<<<nblocks, BLOCK_T, 0, stream>>>(h1, erow, ecol, evl, out, E);
}